// MPS_FNN_73005854097731
// MI455X (gfx1250) — compile-verified
//
#include <hip/hip_runtime.h>
#include <math.h>

typedef float v2f __attribute__((ext_vector_type(2)));
typedef float v8f __attribute__((ext_vector_type(8)));

#define B_SZ   2048
#define FSITES 64
#define CHI    128
#define DD     17
#define NFREQ  8
#define KTOT   (CHI * DD)   // 2176

#define S_STRIDE 130   // s_lds row stride (banks: 2*row + i, even spacing)
#define F_STRIDE 17
#define OL_STRIDE 130  // left out staging [16][128]
#define OR_STRIDE 18   // right out staging [128][16]

__device__ __forceinline__ float feat_val(int p, float xv) {
  if (p == 0) return 1.0f;
  if (p <= NFREQ) return __sinf((float)p * xv);
  return __cosf((float)(p - NFREQ) * xv);
}

// ---------------- site 0: ls17[b,j] = sum_p A0[p,j] * f[b,0,p]; renorm ----------------
__global__ __launch_bounds__(256) void site0_kernel(const float* __restrict__ A0,
                                                    const float* __restrict__ x,
                                                    const float* __restrict__ fminp,
                                                    float* __restrict__ t17,
                                                    float* __restrict__ logn) {
  int b = blockIdx.x * 256 + threadIdx.x;
  if (b >= B_SZ) return;
  float xv = x[b * FSITES + 0] * fminp[0];
  float f[DD];
  for (int p = 0; p < DD; ++p) f[p] = feat_val(p, xv);
  float o[DD], ss = 0.0f;
  for (int j = 0; j < DD; ++j) {
    float a = 0.0f;
    for (int p = 0; p < DD; ++p) a += A0[p * DD + j] * f[p];
    o[j] = a; ss += a * a;
  }
  float n = sqrtf(ss) + 1e-8f, inv = 1.0f / n;
  for (int j = 0; j < DD; ++j) t17[b * DD + j] = o[j] * inv;
  logn[b] = logf(n);  // first contribution: initializes poisoned ws
}

// ---------------- site 63: rs17[b,l] = sum_p A63[l,p] * f[b,63,p]; renorm ----------------
__global__ __launch_bounds__(256) void site63_kernel(const float* __restrict__ A63,
                                                     const float* __restrict__ x,
                                                     const float* __restrict__ fminp,
                                                     float* __restrict__ t17,
                                                     float* __restrict__ logn) {
  int b = blockIdx.x * 256 + threadIdx.x;
  if (b >= B_SZ) return;
  float xv = x[b * FSITES + 63] * fminp[0];
  float f[DD];
  for (int p = 0; p < DD; ++p) f[p] = feat_val(p, xv);
  float o[DD], ss = 0.0f;
  for (int l = 0; l < DD; ++l) {
    float a = 0.0f;
    for (int p = 0; p < DD; ++p) a += A63[l * DD + p] * f[p];
    o[l] = a; ss += a * a;
  }
  float n = sqrtf(ss) + 1e-8f, inv = 1.0f / n;
  for (int l = 0; l < DD; ++l) t17[b * DD + l] = o[l] * inv;
  logn[b] += logf(n);
}

// ---------------- site 1: ls[b,j<128] = sum_{i,p} ls17[b,i] A1[i,p,j] f[b,1,p] ----------------
__global__ __launch_bounds__(128) void site1_kernel(const float* __restrict__ A1,
                                                    const float* __restrict__ x,
                                                    const float* __restrict__ fminp,
                                                    const float* __restrict__ t17,
                                                    float* __restrict__ ls,
                                                    float* __restrict__ logn) {
  __shared__ float fsh[DD], lsh[DD], red[128], invn;
  int b = blockIdx.x, t = threadIdx.x;
  if (t < DD) lsh[t] = t17[b * DD + t];
  if (t >= 32 && t < 32 + DD) {
    int p = t - 32;
    fsh[p] = feat_val(p, x[b * FSITES + 1] * fminp[0]);
  }
  __syncthreads();
  float a = 0.0f;
  for (int i = 0; i < DD; ++i) {
    float li = lsh[i];
    for (int p = 0; p < DD; ++p) a += li * fsh[p] * A1[(i * DD + p) * CHI + t];
  }
  red[t] = a * a; __syncthreads();
  for (int o = 64; o > 0; o >>= 1) { if (t < o) red[t] += red[t + o]; __syncthreads(); }
  if (t == 0) { float n = sqrtf(red[0]) + 1e-8f; invn = 1.0f / n; logn[b] += logf(n); }
  __syncthreads();
  ls[b * CHI + t] = a * invn;
}

// ---------------- site 62: rs[b,l<128] = sum_{p,r} A62[l,p,r] f[b,62,p] rs17[b,r] ----------------
__global__ __launch_bounds__(128) void site62_kernel(const float* __restrict__ A62,
                                                     const float* __restrict__ x,
                                                     const float* __restrict__ fminp,
                                                     const float* __restrict__ t17,
                                                     float* __restrict__ rs,
                                                     float* __restrict__ logn) {
  __shared__ float fsh[DD], rsh[DD], red[128], invn;
  int b = blockIdx.x, t = threadIdx.x;
  if (t < DD) rsh[t] = t17[b * DD + t];
  if (t >= 32 && t < 32 + DD) {
    int p = t - 32;
    fsh[p] = feat_val(p, x[b * FSITES + 62] * fminp[0]);
  }
  __syncthreads();
  float a = 0.0f;
  const float* row = A62 + (size_t)t * DD * DD;  // l = t
  for (int p = 0; p < DD; ++p) {
    float fp = fsh[p];
    for (int r = 0; r < DD; ++r) a += row[p * DD + r] * fp * rsh[r];
  }
  red[t] = a * a; __syncthreads();
  for (int o = 64; o > 0; o >>= 1) { if (t < o) red[t] += red[t + o]; __syncthreads(); }
  if (t == 0) { float n = sqrtf(red[0]) + 1e-8f; invn = 1.0f / n; logn[b] += logf(n); }
  __syncthreads();
  rs[b * CHI + t] = a * invn;
}

// ---------------- generic LEFT site (WMMA f32 16x16x4) ----------------
// D[b,j] = sum_k L[b,k] * W[k,j],  k=(i,p), L[b,k]=s[b,i]*f[b,p], W row-contiguous in j.
__global__ __launch_bounds__(128) void left_site_kernel(const float* __restrict__ A,  // [CHI][DD][CHI]
                                                        const float* __restrict__ x,
                                                        const float* __restrict__ fminp,
                                                        float* __restrict__ s,       // [B][CHI] in/out
                                                        float* __restrict__ logn,
                                                        int site) {
  __shared__ float s_lds[16 * S_STRIDE];
  __shared__ float f_lds[16 * F_STRIDE];
  __shared__ float o_lds[16 * OL_STRIDE];
  __shared__ float n_lds[16];

  int t = threadIdx.x;
  int b0 = blockIdx.x * 16;
  float fmin = fminp[0];

  for (int r = 0; r < 16; ++r)
    s_lds[r * S_STRIDE + t] = s[(size_t)(b0 + r) * CHI + t];
  for (int idx = t; idx < 16 * DD; idx += 128) {
    int r = idx / DD, p = idx % DD;
    f_lds[r * F_STRIDE + p] = feat_val(p, x[(b0 + r) * FSITES + site] * fmin);
  }
  __syncthreads();

  int lane = t & 31;
  int wave = t >> 5;               // 0..3
  int col  = lane & 15;            // A: M row (batch), B: N col (j)
  int koff = (lane >> 4) << 1;     // 0 or 2
  int n0   = wave * 32;            // two N tiles: n0, n0+16
  int rhi  = (lane >> 4) * 8;

  v8f acc0 = {}; v8f acc1 = {};
  for (int p = 0; p < DD; ++p) {
    float fp = f_lds[col * F_STRIDE + p];
    const float* Wp = A + (size_t)p * CHI;   // W[(i*DD+p)*CHI + j]
    #pragma unroll 4
    for (int i0 = 0; i0 < CHI; i0 += 4) {
      int ka = i0 + koff;
      v2f a;
      a.x = s_lds[col * S_STRIDE + ka] * fp;
      a.y = s_lds[col * S_STRIDE + ka + 1] * fp;
      const float* Wk0 = Wp + (size_t)ka * DD * CHI;
      const float* Wk1 = Wp + (size_t)(ka + 1) * DD * CHI;
      v2f b0v, b1v;
      b0v.x = Wk0[n0 + col];       b0v.y = Wk1[n0 + col];
      b1v.x = Wk0[n0 + 16 + col];  b1v.y = Wk1[n0 + 16 + col];
      acc0 = __builtin_amdgcn_wmma_f32_16x16x4_f32(false, a, false, b0v, (short)0, acc0, false, false);
      acc1 = __builtin_amdgcn_wmma_f32_16x16x4_f32(false, a, false, b1v, (short)0, acc1, false, false);
    }
  }
  // D layout: VGPR r -> row (r + 8*(lane>=16)), col = lane&15
  for (int r = 0; r < 8; ++r) {
    o_lds[(r + rhi) * OL_STRIDE + n0 + col]      = acc0[r];
    o_lds[(r + rhi) * OL_STRIDE + n0 + 16 + col] = acc1[r];
  }
  __syncthreads();
  if (t < 16) {
    float ss = 0.0f;
    for (int j = 0; j < CHI; ++j) { float v = o_lds[t * OL_STRIDE + j]; ss += v * v; }
    float n = sqrtf(ss) + 1e-8f;
    n_lds[t] = 1.0f / n;
    logn[b0 + t] += logf(n);
  }
  __syncthreads();
  for (int r = 0; r < 16; ++r)
    s[(size_t)(b0 + r) * CHI + t] = o_lds[r * OL_STRIDE + t] * n_lds[r];
}

// ---------------- generic RIGHT site (WMMA f32 16x16x4, transposed product) ----------------
// s_new[b,l] = sum_{p,r} A[l,p,r] f[b,p] s[b,r].  Compute D^T[l,b]: A-matrix = weights
// (per-lane row l, contiguous K=r reads), B-matrix = (f*s)^T from LDS.
__global__ __launch_bounds__(128) void right_site_kernel(const float* __restrict__ A, // [CHI][DD][CHI] (l,p,r)
                                                         const float* __restrict__ x,
                                                         const float* __restrict__ fminp,
                                                         float* __restrict__ s,      // [B][CHI] in/out
                                                         float* __restrict__ logn,
                                                         int site) {
  __shared__ float s_lds[16 * S_STRIDE];
  __shared__ float f_lds[16 * F_STRIDE];
  __shared__ float o_lds[CHI * OR_STRIDE];  // [l][b]
  __shared__ float n_lds[16];

  int t = threadIdx.x;
  int b0 = blockIdx.x * 16;
  float fmin = fminp[0];

  for (int r = 0; r < 16; ++r)
    s_lds[r * S_STRIDE + t] = s[(size_t)(b0 + r) * CHI + t];
  for (int idx = t; idx < 16 * DD; idx += 128) {
    int r = idx / DD, p = idx % DD;
    f_lds[r * F_STRIDE + p] = feat_val(p, x[(b0 + r) * FSITES + site] * fmin);
  }
  __syncthreads();

  int lane = t & 31;
  int wave = t >> 5;
  int col  = lane & 15;            // A: M row (l), B: N col (batch)
  int koff = (lane >> 4) << 1;
  int l0   = wave * 32;            // two M tiles of l: l0, l0+16
  int rhi  = (lane >> 4) * 8;

  v8f acc0 = {}; v8f acc1 = {};
  for (int p = 0; p < DD; ++p) {
    float fb = f_lds[col * F_STRIDE + p];
    const float* Al0 = A + (size_t)(l0 + col) * KTOT + (size_t)p * CHI;
    const float* Al1 = A + (size_t)(l0 + 16 + col) * KTOT + (size_t)p * CHI;
    #pragma unroll 4
    for (int i0 = 0; i0 < CHI; i0 += 4) {
      int ka = i0 + koff;
      v2f aw0, aw1, bl;
      aw0.x = Al0[ka];  aw0.y = Al0[ka + 1];   // contiguous -> b64, streams cachelines
      aw1.x = Al1[ka];  aw1.y = Al1[ka + 1];
      bl.x = s_lds[col * S_STRIDE + ka] * fb;
      bl.y = s_lds[col * S_STRIDE + ka + 1] * fb;
      acc0 = __builtin_amdgcn_wmma_f32_16x16x4_f32(false, aw0, false, bl, (short)0, acc0, false, false);
      acc1 = __builtin_amdgcn_wmma_f32_16x16x4_f32(false, aw1, false, bl, (short)0, acc1, false, false);
    }
  }
  for (int r = 0; r < 8; ++r) {
    o_lds[(l0 + rhi + r) * OR_STRIDE + col]      = acc0[r];
    o_lds[(l0 + 16 + rhi + r) * OR_STRIDE + col] = acc1[r];
  }
  __syncthreads();
  if (t < 16) {
    float ss = 0.0f;
    for (int l = 0; l < CHI; ++l) { float v = o_lds[l * OR_STRIDE + t]; ss += v * v; }
    float n = sqrtf(ss) + 1e-8f;
    n_lds[t] = 1.0f / n;
    logn[b0 + t] += logf(n);
  }
  __syncthreads();
  for (int br = 0; br < 16; ++br)
    s[(size_t)(b0 + br) * CHI + t] = o_lds[t * OR_STRIDE + br] * n_lds[br];
}

// ---------------- center ----------------
__global__ __launch_bounds__(128) void center_kernel(const float* __restrict__ Ac,  // [CHI][DD][CHI][1]
                                                     const float* __restrict__ x,
                                                     const float* __restrict__ fminp,
                                                     const float* __restrict__ ls,
                                                     const float* __restrict__ rs,
                                                     const float* __restrict__ logn,
                                                     float* __restrict__ out) {
  __shared__ float lsh[CHI], rsh[CHI], fsh[DD], red[128];
  int b = blockIdx.x, t = threadIdx.x;
  lsh[t] = ls[b * CHI + t];
  rsh[t] = rs[b * CHI + t];
  if (t < DD) fsh[t] = feat_val(t, x[b * FSITES + 31] * fminp[0]);
  __syncthreads();
  float a = 0.0f;
  for (int rep = 0; rep < DD; ++rep) {
    int idx = rep * 128 + t;          // (i*DD + p), 2176 total
    int i = idx / DD, p = idx % DD;
    const float* row = Ac + (size_t)idx * CHI;
    float inner = 0.0f;
    for (int r = 0; r < CHI; ++r) inner += row[r] * rsh[r];
    a += lsh[i] * fsh[p] * inner;
  }
  red[t] = a; __syncthreads();
  for (int o = 64; o > 0; o >>= 1) { if (t < o) red[t] += red[t + o]; __syncthreads(); }
  if (t == 0) {
    float o_ = red[0];
    float n = fabsf(o_) + 1e-8f;
    out[b] = (o_ / n) * expf(logn[b] + logf(n));
  }
}

extern "C" void kernel_launch(void* const* d_in, const int* in_sizes, int n_in,
                              void* d_out, int out_size, void* d_ws, size_t ws_size,
                              hipStream_t stream) {
  (void)in_sizes; (void)n_in; (void)out_size; (void)ws_size;
  const float* x    = (const float*)d_in[0];
  const float* fmin = (const float*)d_in[1];
  const float* A0   = (const float*)d_in[2];
  const float* A1   = (const float*)d_in[3];
  const float* AL   = (const float*)d_in[4];
  const float* AC   = (const float*)d_in[5];
  const float* AR   = (const float*)d_in[6];
  const float* A62  = (const float*)d_in[7];
  const float* A63  = (const float*)d_in[8];

  float* ws   = (float*)d_ws;
  float* ls   = ws;                                  // B*128
  float* rs   = ws + (size_t)B_SZ * CHI;             // B*128
  float* t17  = ws + (size_t)2 * B_SZ * CHI;         // B*17 (padded to B*20)
  float* logn = ws + (size_t)2 * B_SZ * CHI + (size_t)B_SZ * 20;
  float* out  = (float*)d_out;

  // left sweep
  site0_kernel<<<B_SZ / 256, 256, 0, stream>>>(A0, x, fmin, t17, logn);
  site1_kernel<<<B_SZ, 128, 0, stream>>>(A1, x, fmin, t17, ls, logn);
  for (int s = 2; s <= 30; ++s)
    left_site_kernel<<<B_SZ / 16, 128, 0, stream>>>(AL + (size_t)(s - 2) * CHI * DD * CHI,
                                                    x, fmin, ls, logn, s);
  // right sweep
  site63_kernel<<<B_SZ / 256, 256, 0, stream>>>(A63, x, fmin, t17, logn);
  site62_kernel<<<B_SZ, 128, 0, stream>>>(A62, x, fmin, t17, rs, logn);
  for (int s = 61; s >= 32; --s)
    right_site_kernel<<<B_SZ / 16, 128, 0, stream>>>(AR + (size_t)(s - 32) * CHI * DD * CHI,
                                                     x, fmin, rs, logn, s);
  // center
  center_kernel<<<B_SZ, 128, 0, stream>>>(AC, x, fmin, ls, rs, logn, out);
}